// MyEmbedding_13907104104670
// MI455X (gfx1250) — compile-verified
//
#include <hip/hip_runtime.h>

typedef __attribute__((ext_vector_type(2))) float v2f;
typedef __attribute__((ext_vector_type(8))) float v8f;

#define GDIM   300   // glove embedding dim
#define KSTEPS 75    // 300 / 4 (K per V_WMMA_F32_16X16X4_F32)
#define ODIM   64    // output dim

// One wave computes a 16-token x 64-dim output tile:
//   out[16,64] += A[16,300] (gathered glove rows) x B[300,64] (W^T, staged in LDS)
// using V_WMMA_F32_16X16X4_F32, then overwrites flagged rows with vocab rows.
__global__ __launch_bounds__(256) void MyEmbedding_kernel(
    const float* __restrict__ glove,   // [400000, 300]
    const float* __restrict__ w,       // [64, 300] row-major
    const float* __restrict__ vocab,   // [12, 64]
    const int*   __restrict__ flags,   // [L]
    const int*   __restrict__ gidx,    // [L]
    const int*   __restrict__ vidx,    // [L]
    float*       __restrict__ out,     // [L, 64]
    int L)
{
    // LDS layout: lds_w[kpair][d] = { w[d][2*kpair], w[d][2*kpair+1] },
    // kpair in [0,150), d in [0,64). 150*64*8 = 76800 B (< 320 KB/WGP).
    // B fragment for k-step k, tile n: lane l reads lds_w[2k + (l>=16)][n*16 + (l&15)]
    // -> per-lane float2 = (K even, K odd) at the lane's K offset, exactly the
    //    4x16 B operand striping (VGPR0: K0/K2 halves, VGPR1: K1/K3 halves).
    __shared__ v2f lds_w[150 * 64];

    const int tid = threadIdx.x;

    // Cooperative stage of W into LDS: linear (coalesced) global reads,
    // scattered LDS writes.
    for (int j = tid; j < ODIM * GDIM; j += 256) {
        int d = j / GDIM;
        int g = j - d * GDIM;
        ((float*)lds_w)[(g >> 1) * 128 + d * 2 + (g & 1)] = w[j];
    }
    __syncthreads();

    const int lane = tid & 31;
    const int half = lane >> 4;        // 0: lanes 0-15, 1: lanes 16-31
    const int m16  = lane & 15;
    const int waveId   = (blockIdx.x * blockDim.x + tid) >> 5;
    const int tok_base = waveId * 16;
    if (tok_base >= L) return;         // whole-wave uniform exit (EXEC stays all-1s)

    // Per-lane token metadata (token = tok_base + (lane&15), duplicated per half)
    int tok = tok_base + m16;
    if (tok >= L) tok = L - 1;         // clamp keeps EXEC full for WMMA
    const int flag = flags[tok];
    const int gi   = (flag == 0) ? gidx[tok] : 0;  // flagged -> row 0 (zeroed), halves gather bytes
    const int vi   = vidx[tok];

    // A gather stream: lane reads float2 at glove[gi*300 + 4k + 2*half].
    const float* arow = glove + (size_t)gi * GDIM + 2 * half;
    __builtin_prefetch(arow, 0, 0);    // global_prefetch_b8 warms first line of the row

    // Per-lane B element offset within one k-step (in float2 elements).
    const int boff = half * 64 + m16;

    v8f acc0 = {}, acc1 = {}, acc2 = {}, acc3 = {};

    for (int k = 0; k < KSTEPS; ++k) {
        v2f a = *(const v2f*)(arow + 4 * k);          // global_load_b64 (gather)
        const v2f* bk = lds_w + k * 128 + boff;       // ds_load_b64 x4, conflict-free
        v2f b0 = bk[0];
        v2f b1 = bk[16];
        v2f b2 = bk[32];
        v2f b3 = bk[48];
        acc0 = __builtin_amdgcn_wmma_f32_16x16x4_f32(false, a, false, b0, (short)0, acc0, false, false);
        acc1 = __builtin_amdgcn_wmma_f32_16x16x4_f32(false, a, false, b1, (short)0, acc1, false, false);
        acc2 = __builtin_amdgcn_wmma_f32_16x16x4_f32(false, a, false, b2, (short)0, acc2, false, false);
        acc3 = __builtin_amdgcn_wmma_f32_16x16x4_f32(false, a, false, b3, (short)0, acc3, false, false);
    }

    // C/D layout: VGPR r <-> token m = r + 8*half, column d = n*16 + (lane&15).
    // Flag select: flagged tokens take the vocab row instead of the GEMM result.
    #pragma unroll
    for (int r = 0; r < 8; ++r) {
        const int m = r + half * 8;
        const int t = tok_base + m;
        if (t >= L) continue;
        const int f  = __shfl(flag, m, 32);
        const int v  = __shfl(vi,   m, 32);
        const float* vrow = vocab + v * ODIM;         // 3 KB table, cache-resident
        float* orow = out + (size_t)t * ODIM + m16;
        float g0 = acc0[r], g1 = acc1[r], g2 = acc2[r], g3 = acc3[r];
        orow[0]  = f ? vrow[m16]      : g0;
        orow[16] = f ? vrow[16 + m16] : g1;
        orow[32] = f ? vrow[32 + m16] : g2;
        orow[48] = f ? vrow[48 + m16] : g3;
    }
}

extern "C" void kernel_launch(void* const* d_in, const int* in_sizes, int n_in,
                              void* d_out, int out_size, void* d_ws, size_t ws_size,
                              hipStream_t stream) {
    const float* glove = (const float*)d_in[0];
    const float* w     = (const float*)d_in[1];
    const float* vocab = (const float*)d_in[2];
    const int*   flags = (const int*)d_in[3];
    const int*   gidx  = (const int*)d_in[4];
    const int*   vidx  = (const int*)d_in[5];
    float* out = (float*)d_out;

    const int L = in_sizes[3];                 // 524288
    const int tokens_per_block = 128;          // 8 wave32 x 16 tokens
    const int blocks = (L + tokens_per_block - 1) / tokens_per_block;

    MyEmbedding_kernel<<<blocks, 256, 0, stream>>>(glove, w, vocab, flags, gidx, vidx, out, L);
}